// GazeLSTM_48893907698268
// MI455X (gfx1250) — compile-verified
//
#include <hip/hip_runtime.h>
#include <hip/hip_bf16.h>
#include <math.h>

// GazeLSTM for MI455X (gfx1250).
// Memory-bound (~100 MB traffic -> ~4.3us floor @ 23.3 TB/s). One lane per
// batch element streams its R slice; the hidden-size-3 LSTM gate GEMM is done
// with V_WMMA_F32_16X16X4_F32 (exact fp32), 16 batch rows per WMMA.

typedef __attribute__((ext_vector_type(2))) float v2f;
typedef __attribute__((ext_vector_type(8))) float v8f;

#define NT   128
#define HALF 64

__device__ __forceinline__ float sigmoid_(float x) {
    return 1.0f / (1.0f + __expf(-x));
}

__global__ __launch_bounds__(32) void gaze_lstm_kernel(
    const float* __restrict__ dirp,
    const float* __restrict__ R,
    const float* __restrict__ W_ih,
    const float* __restrict__ W_hh,
    const float* __restrict__ b_ih,
    const float* __restrict__ b_hh,
    float* __restrict__ out)
{
    // xs[t][lane][k] : input sequence to the LSTM (dirs_tm), t = 0..HALF
    __shared__ float xs[HALF + 1][32][3];
    __shared__ float gate_s[32 * 13];   // stride 13 -> bank-conflict free

    const int  lane = threadIdx.x;           // 0..31 (wave32)
    const int  b    = blockIdx.x * 32 + lane;
    const bool lo   = lane < 16;
    const int  n    = lane & 15;             // gate column this lane carries in B

    // ---- WMMA B operands (constant over time). B is 4x16 (K x N), f32. ----
    // Assumed layout mirrors the documented 16x4 A layout:
    //   VGPR0: lanes 0-15 = row K=0, lanes 16-31 = row K=2
    //   VGPR1: lanes 0-15 = row K=1, lanes 16-31 = row K=3
    // Row K=3 of Bx carries the fused bias (A's K=3 element is 1.0);
    // row K=3 of Bh is zero (A_h's K=3 element is 0.0).
    const int   nn  = (n < 12) ? n : 0;
    const float msk = (n < 12) ? 1.0f : 0.0f;
    const float wi0 = msk * W_ih[nn * 3 + 0];
    const float wi1 = msk * W_ih[nn * 3 + 1];
    const float wi2 = msk * W_ih[nn * 3 + 2];
    const float wg0 = msk * W_hh[nn * 3 + 0];
    const float wg1 = msk * W_hh[nn * 3 + 1];
    const float wg2 = msk * W_hh[nn * 3 + 2];
    const float bs  = msk * (b_ih[nn] + b_hh[nn]);

    v2f Bx, Bh;
    Bx.x = lo ? wi0 : wi2;
    Bx.y = lo ? wi1 : bs;
    Bh.x = lo ? wg0 : wg2;
    Bh.y = lo ? wg1 : 0.0f;

    // ------------- Phase 1: rotation chains (HBM streaming) ---------------
    const float dx = dirp[b * 3 + 0];
    const float dy = dirp[b * 3 + 1];
    const float dz = dirp[b * 3 + 2];

    xs[HALF][lane][0] = dx; xs[HALF][lane][1] = dy; xs[HALF][lane][2] = dz;

    const float* Rb = R + (size_t)b * NT * 9;
    float*       ob = out + (size_t)b * NT * 3;

    // Backward chain: dirs_tm[j] = R[HALF-1-j]^T @ ... @ R[HALF-1]^T @ dir
    float v0 = dx, v1 = dy, v2 = dz;
    for (int j = 0; j < HALF; ++j) {
        const float* M = Rb + (size_t)(HALF - 1 - j) * 9;
        if (j + 4 < HALF) __builtin_prefetch(Rb + (size_t)(HALF - 5 - j) * 9, 0, 1);
        const float m00 = M[0], m01 = M[1], m02 = M[2];
        const float m10 = M[3], m11 = M[4], m12 = M[5];
        const float m20 = M[6], m21 = M[7], m22 = M[8];
        const float t0 = m00 * v0 + m10 * v1 + m20 * v2;   // M^T v
        const float t1 = m01 * v0 + m11 * v1 + m21 * v2;
        const float t2 = m02 * v0 + m12 * v1 + m22 * v2;
        v0 = t0; v1 = t1; v2 = t2;
        ob[j * 3 + 0] = v0; ob[j * 3 + 1] = v1; ob[j * 3 + 2] = v2;
        xs[j][lane][0] = v0; xs[j][lane][1] = v1; xs[j][lane][2] = v2;
    }

    // Forward chain: dirs_tm[HALF+1+m] = R[HALF+m] @ ... @ R[HALF] @ dir
    v0 = dx; v1 = dy; v2 = dz;
    for (int m = 0; m < NT - 1 - HALF; ++m) {
        const float* M = Rb + (size_t)(HALF + m) * 9;
        if (m + 4 < NT - 1 - HALF) __builtin_prefetch(Rb + (size_t)(HALF + m + 4) * 9, 0, 1);
        const float m00 = M[0], m01 = M[1], m02 = M[2];
        const float m10 = M[3], m11 = M[4], m12 = M[5];
        const float m20 = M[6], m21 = M[7], m22 = M[8];
        const float t0 = m00 * v0 + m01 * v1 + m02 * v2;   // M v
        const float t1 = m10 * v0 + m11 * v1 + m12 * v2;
        const float t2 = m20 * v0 + m21 * v1 + m22 * v2;
        v0 = t0; v1 = t1; v2 = t2;
        const int tt = HALF + 1 + m;
        ob[tt * 3 + 0] = v0; ob[tt * 3 + 1] = v1; ob[tt * 3 + 2] = v2;
    }

    // ------------- Phase 2: LSTM t = 0..HALF via f32 WMMA -----------------
    // gates[32x12] per wave = [x,1](32x4) @ [W_ih^T;bias](4x16)
    //                       + [h,0](32x4) @ [W_hh^T;0 ](4x16)
    float h0 = 0.f, h1 = 0.f, h2 = 0.f;
    float c0 = 0.f, c1 = 0.f, c2 = 0.f;

    for (int t = 0; t <= HALF; ++t) {
        const float x0 = xs[t][lane][0];
        const float x1 = xs[t][lane][1];
        const float x2 = xs[t][lane][2];
        const float p0 = __shfl_xor(x0, 16);
        const float p1 = __shfl_xor(x1, 16);
        const float p2 = __shfl_xor(x2, 16);
        const float q0 = __shfl_xor(h0, 16);
        const float q1 = __shfl_xor(h1, 16);
        const float q2 = __shfl_xor(h2, 16);

        // A operand layout (16x4 f32): lanes 0-15 hold K0/K1 of row M=lane,
        // lanes 16-31 hold K2/K3 of row M=lane-16.
        v2f A0, A1, H0, H1;
        A0.x = lo ? x0 : p2;  A0.y = lo ? x1 : 1.0f;   // group 0: batch rows 0..15
        A1.x = lo ? p0 : x2;  A1.y = lo ? p1 : 1.0f;   // group 1: batch rows 16..31
        H0.x = lo ? h0 : q2;  H0.y = lo ? h1 : 0.0f;
        H1.x = lo ? q0 : h2;  H1.y = lo ? q1 : 0.0f;

        v8f acc0 = {};
        v8f acc1 = {};
        acc0 = __builtin_amdgcn_wmma_f32_16x16x4_f32(false, H0, false, Bh,
                                                     (short)0, acc0, false, false);
        acc0 = __builtin_amdgcn_wmma_f32_16x16x4_f32(false, A0, false, Bx,
                                                     (short)0, acc0, false, false);
        acc1 = __builtin_amdgcn_wmma_f32_16x16x4_f32(false, H1, false, Bh,
                                                     (short)0, acc1, false, false);
        acc1 = __builtin_amdgcn_wmma_f32_16x16x4_f32(false, A1, false, Bx,
                                                     (short)0, acc1, false, false);

        // D layout (16x16 f32): VGPR r holds M=r (lanes 0-15, N=lane) and
        // M=8+r (lanes 16-31, N=lane-16). Transpose via LDS to per-lane rows.
        if (n < 12) {
            const int mb = lo ? 0 : 8;
#pragma unroll
            for (int r = 0; r < 8; ++r) {
                gate_s[(mb + r) * 13 + n]      = acc0[r];
                gate_s[(16 + mb + r) * 13 + n] = acc1[r];
            }
        }
        __syncthreads();

        const float* g = &gate_s[lane * 13];
        const float i0 = sigmoid_(g[0]), i1 = sigmoid_(g[1]), i2 = sigmoid_(g[2]);
        const float f0 = sigmoid_(g[3]), f1 = sigmoid_(g[4]), f2 = sigmoid_(g[5]);
        const float z0 = tanhf(g[6]),    z1 = tanhf(g[7]),    z2 = tanhf(g[8]);
        const float o0 = sigmoid_(g[9]), o1 = sigmoid_(g[10]), o2 = sigmoid_(g[11]);
        c0 = f0 * c0 + i0 * z0;
        c1 = f1 * c1 + i1 * z1;
        c2 = f2 * c2 + i2 * z2;
        h0 = o0 * tanhf(c0);
        h1 = o1 * tanhf(c1);
        h2 = o2 * tanhf(c2);
        __syncthreads();
    }

    // out[:, HALF, :] = h_seq[HALF]
    ob[HALF * 3 + 0] = h0;
    ob[HALF * 3 + 1] = h1;
    ob[HALF * 3 + 2] = h2;
}

extern "C" void kernel_launch(void* const* d_in, const int* in_sizes, int n_in,
                              void* d_out, int out_size, void* d_ws, size_t ws_size,
                              hipStream_t stream) {
    const float* dirp = (const float*)d_in[0];
    const float* R    = (const float*)d_in[1];
    const float* W_ih = (const float*)d_in[2];
    const float* W_hh = (const float*)d_in[3];
    const float* b_ih = (const float*)d_in[4];
    const float* b_hh = (const float*)d_in[5];
    float* out = (float*)d_out;

    const int Bsz = in_sizes[0] / 3;       // 16384
    dim3 grid(Bsz / 32), block(32);        // 512 single-wave workgroups (wave32)
    gaze_lstm_kernel<<<grid, block, 0, stream>>>(dirp, R, W_ih, W_hh, b_ih, b_hh, out);
}